// PrimitiveCollisionCost_63599875719353
// MI455X (gfx1250) — compile-verified
//
#include <hip/hip_runtime.h>

typedef __attribute__((ext_vector_type(2))) float v2f;
typedef __attribute__((ext_vector_type(8))) float v8f;

#define NB 16
#define NH 64
#define NL 12
#define NS 8
#define NW 256
#define WAVES_PER_BLOCK 6              // NL / 2 links-per-wave
#define BLOCK_THREADS (WAVES_PER_BLOCK * 32)

#define DIST_THRESHOLD 0.1f
#define CLIP_MAX 0.2f
#define INV_VIOL_SCALE 4.0f            // 1 / 0.25

__global__ __launch_bounds__(BLOCK_THREADS)
void PrimitiveCollisionCost_kernel(const float* __restrict__ pos,    // (1024,12,3)
                                   const float* __restrict__ rot,    // (1024,12,3,3)
                                   const float* __restrict__ lsph,   // (12,8,4) xyz,r
                                   const float* __restrict__ wsph,   // (256,4)  xyz,r
                                   const float* __restrict__ weight, // scalar
                                   float* __restrict__ out)          // (1024,)
{
    __shared__ float4 s_world[NW];
    __shared__ float  s_wsum[WAVES_PER_BLOCK];

    const int tid  = threadIdx.x;
    const int n    = blockIdx.x;       // flattened (b*h) pose index
    const int w    = tid >> 5;         // wave id 0..5
    const int lane = tid & 31;
    const int hi   = lane >> 4;        // half of the wave (K/row-half select)
    const int m    = lane & 15;        // A-row / B-col handled by this lane

    // Stage all world spheres into LDS once per block (4 KB).
    {
        const float4* wp = (const float4*)wsph;
        for (int i = tid; i < NW; i += BLOCK_THREADS) s_world[i] = wp[i];
    }
    __syncthreads();

    // ---- A operand: 16 robot-sphere centers = 2 links x 8 spheres.
    // Row m -> link_in_wave = m>>3, sphere s = m&7.
    // Lanes 0-15 : A = {K0=cx,  K1=cy }   for row M = lane
    // Lanes 16-31: A = {K2=cz,  K3=|c|^2} for row M = lane-16
    const int s  = m & 7;
    const int l  = 2 * w + (m >> 3);
    const int li = n * NL + l;

    const float* R = rot + (size_t)li * 9;
    const float* P = pos + (size_t)li * 3;
    const float4 sp = ((const float4*)lsph)[l * NS + s];

    const float cx = R[0]*sp.x + R[1]*sp.y + R[2]*sp.z + P[0];
    const float cy = R[3]*sp.x + R[4]*sp.y + R[5]*sp.z + P[1];
    const float cz = R[6]*sp.x + R[7]*sp.y + R[8]*sp.z + P[2];
    const float c2 = cx*cx + cy*cy + cz*cz;

    v2f a;
    a.x = hi ? cz : cx;
    a.y = hi ? c2 : cy;

    // Running per-lane maxima of (w_r - dist), one per D register (row).
    float rmax[8];
    #pragma unroll
    for (int r = 0; r < 8; ++r) rmax[r] = -3.0e38f;

    #pragma unroll
    for (int t = 0; t < NW / 16; ++t) {
        // ---- B operand: 16 world spheres; this lane's column = m.
        // Lanes 0-15 : B = {K0=-2wx, K1=-2wy}
        // Lanes 16-31: B = {K2=-2wz, K3= 1  }
        const float4 ws = s_world[t * 16 + m];
        v2f b;
        b.x = hi ? (-2.0f * ws.z) : (-2.0f * ws.x);
        b.y = hi ? 1.0f           : (-2.0f * ws.y);
        const float w2 = ws.x*ws.x + ws.y*ws.y + ws.z*ws.z; // col-m value, both halves
        const float wr = ws.w;

        v8f c = {0.f, 0.f, 0.f, 0.f, 0.f, 0.f, 0.f, 0.f};
        // D[row][col] = |c_row|^2 - 2 * c_row . w_col
        c = __builtin_amdgcn_wmma_f32_16x16x4_f32(
                /*neg_a=*/false, a, /*neg_b=*/false, b,
                /*c_mod=*/(short)0, c, /*reuse_a=*/false, /*reuse_b=*/false);

        // Three independent passes so the 8 v_sqrt_f32 (TRANS pipe) pipeline
        // back-to-back instead of serializing through one register with
        // v_nop hazard slots between each sqrt and its consumer.
        float d2[8];
        #pragma unroll
        for (int r = 0; r < 8; ++r)
            d2[r] = fmaxf(c[r] + w2, 0.0f);           // ||c - w||^2 (clamp fp error)

        float sq[8];
        #pragma unroll
        for (int r = 0; r < 8; ++r)
            sq[r] = __builtin_amdgcn_sqrtf(d2[r]);    // raw v_sqrt_f32 (well-scaled)

        #pragma unroll
        for (int r = 0; r < 8; ++r)
            rmax[r] = fmaxf(rmax[r], wr - sq[r]);
    }

    // Max over columns: xor-shuffle reduction within each 16-lane half.
    #pragma unroll
    for (int r = 0; r < 8; ++r) {
        float v = rmax[r];
        v = fmaxf(v, __shfl_xor(v, 1, 32));
        v = fmaxf(v, __shfl_xor(v, 2, 32));
        v = fmaxf(v, __shfl_xor(v, 4, 32));
        v = fmaxf(v, __shfl_xor(v, 8, 32));
        rmax[r] = v;
    }

    // Rows r + 8*hi belong to link (2w + hi), sphere s = r.
    // Add robot radii, take max over the 8 spheres.
    const int lh = 2 * w + hi;
    float linkmax = -3.0e38f;
    #pragma unroll
    for (int r = 0; r < 8; ++r) {
        const float rr = lsph[(lh * NS + r) * 4 + 3];
        linkmax = fmaxf(linkmax, rmax[r] + rr);
    }

    // d = clip(dist + 0.1, 0, 0.2) / 0.25 ; combine this wave's two links.
    float d = fminf(fmaxf(linkmax + DIST_THRESHOLD, 0.0f), CLIP_MAX) * INV_VIOL_SCALE;
    d += __shfl_xor(d, 16, 32);

    if (lane == 0) s_wsum[w] = d;
    __syncthreads();

    if (tid == 0) {
        float tot = 0.0f;
        #pragma unroll
        for (int i = 0; i < WAVES_PER_BLOCK; ++i) tot += s_wsum[i];
        out[n] = weight[0] * tot;
    }
}

extern "C" void kernel_launch(void* const* d_in, const int* in_sizes, int n_in,
                              void* d_out, int out_size, void* d_ws, size_t ws_size,
                              hipStream_t stream) {
    const float* link_pos_seq  = (const float*)d_in[0]; // (16,64,12,3)
    const float* link_rot_seq  = (const float*)d_in[1]; // (16,64,12,3,3)
    const float* link_spheres  = (const float*)d_in[2]; // (12,8,4)
    const float* world_spheres = (const float*)d_in[3]; // (256,4)
    const float* weight        = (const float*)d_in[4]; // scalar
    float* out = (float*)d_out;                         // (16,64) = 1024 floats

    const int n_poses = NB * NH; // 1024 blocks, one pose each
    PrimitiveCollisionCost_kernel<<<n_poses, BLOCK_THREADS, 0, stream>>>(
        link_pos_seq, link_rot_seq, link_spheres, world_spheres, weight, out);
}